// WordnetDGN_9612136808449
// MI455X (gfx1250) — compile-verified
//
#include <hip/hip_runtime.h>
#include <hip/hip_bf16.h>

#define N_NODES 120000
#define N_EDGES 1000000
#define HID 256
#define OUTF 128
#define LN_EPS 1e-12f

typedef float v2f __attribute__((ext_vector_type(2)));
typedef float v8f __attribute__((ext_vector_type(8)));

// ---------------------------------------------------------------------------
// 1) Embedding sum + LayerNorm.  One wave32 per node, 8 columns per lane.
// ---------------------------------------------------------------------------
__global__ __launch_bounds__(256) void embed_ln_kernel(
    const int* __restrict__ x,
    const float* __restrict__ syn, const float* __restrict__ lem,
    const float* __restrict__ pos, const float* __restrict__ sen,
    const float* __restrict__ gamma, const float* __restrict__ beta,
    float* __restrict__ h) {
  const int node = (blockIdx.x * blockDim.x + threadIdx.x) >> 5;
  const int lane = threadIdx.x & 31;
  if (node >= N_NODES) return;

  const int syn_id = x[node * 4 + 0];
  const int pos_id = x[node * 4 + 1];
  const int sen_id = x[node * 4 + 2];
  const int lem_id = x[node * 4 + 3];

  const float4* ps = (const float4*)(syn + (size_t)syn_id * HID);
  const float4* pl = (const float4*)(lem + (size_t)lem_id * HID);
  const float4* pp = (const float4*)(pos + (size_t)pos_id * HID);
  const float4* pe = (const float4*)(sen + (size_t)sen_id * HID);

  const int i0 = lane * 2, i1 = lane * 2 + 1;
  float4 a0 = ps[i0], a1 = ps[i1];
  float4 b0 = pl[i0], b1 = pl[i1];
  float4 c0 = pp[i0], c1 = pp[i1];
  float4 d0 = pe[i0], d1 = pe[i1];

  float4 v0, v1;
  v0.x = a0.x + b0.x + c0.x + d0.x;  v0.y = a0.y + b0.y + c0.y + d0.y;
  v0.z = a0.z + b0.z + c0.z + d0.z;  v0.w = a0.w + b0.w + c0.w + d0.w;
  v1.x = a1.x + b1.x + c1.x + d1.x;  v1.y = a1.y + b1.y + c1.y + d1.y;
  v1.z = a1.z + b1.z + c1.z + d1.z;  v1.w = a1.w + b1.w + c1.w + d1.w;

  float s  = v0.x + v0.y + v0.z + v0.w + v1.x + v1.y + v1.z + v1.w;
  float sq = v0.x*v0.x + v0.y*v0.y + v0.z*v0.z + v0.w*v0.w
           + v1.x*v1.x + v1.y*v1.y + v1.z*v1.z + v1.w*v1.w;
#pragma unroll
  for (int off = 16; off >= 1; off >>= 1) {
    s  += __shfl_xor(s,  off, 32);
    sq += __shfl_xor(sq, off, 32);
  }
  const float mu  = s * (1.0f / HID);
  float var = sq * (1.0f / HID) - mu * mu;
  var = fmaxf(var, 0.0f);
  const float rs = rsqrtf(var + LN_EPS);

  const float4 g0 = ((const float4*)gamma)[i0], g1 = ((const float4*)gamma)[i1];
  const float4 t0 = ((const float4*)beta)[i0],  t1 = ((const float4*)beta)[i1];
  float4 o0, o1;
  o0.x = (v0.x - mu) * rs * g0.x + t0.x;  o0.y = (v0.y - mu) * rs * g0.y + t0.y;
  o0.z = (v0.z - mu) * rs * g0.z + t0.z;  o0.w = (v0.w - mu) * rs * g0.w + t0.w;
  o1.x = (v1.x - mu) * rs * g1.x + t1.x;  o1.y = (v1.y - mu) * rs * g1.y + t1.y;
  o1.z = (v1.z - mu) * rs * g1.z + t1.z;  o1.w = (v1.w - mu) * rs * g1.w + t1.w;

  float4* hp = (float4*)(h + (size_t)node * HID);
  hp[i0] = o0;
  hp[i1] = o1;
}

// ---------------------------------------------------------------------------
// 2) GCN symmetric norm: deg = 1 (self loop) + segment_sum(w, col); dinv = rsqrt
// ---------------------------------------------------------------------------
__global__ void deg_init_kernel(float* __restrict__ deg) {
  int i = blockIdx.x * blockDim.x + threadIdx.x;
  if (i < N_NODES) deg[i] = 1.0f;
}
__global__ void deg_accum_kernel(const int* __restrict__ col,
                                 const float* __restrict__ w,
                                 float* __restrict__ deg) {
  int e = blockIdx.x * blockDim.x + threadIdx.x;
  if (e < N_EDGES) atomicAdd(&deg[col[e]], w[e]);
}
__global__ void dinv_kernel(float* __restrict__ deg) {
  int i = blockIdx.x * blockDim.x + threadIdx.x;
  if (i < N_NODES) {
    float d = deg[i];
    deg[i] = (d > 0.0f) ? rsqrtf(d) : 0.0f;
  }
}

// ---------------------------------------------------------------------------
// 3) Repack W[256,NOUT] into WMMA B-fragment order:
//    fragment (colTile ct, k-step k4) = 32 lanes x v2f, contiguous 256B.
//    lane -> (n = lane&15, koff = lane<16?0:2):
//      Wt[((ct*64 + k4)*64) + lane*2 + {0,1}] = W[(4*k4+koff+{0,1})*NOUT + ct*16 + n]
// ---------------------------------------------------------------------------
template <int NOUT>
__global__ __launch_bounds__(256) void repack_w_kernel(
    const float* __restrict__ W, float* __restrict__ Wt) {
  const int gid = blockIdx.x * blockDim.x + threadIdx.x;  // (NOUT/16)*64*32
  const int lane = gid & 31;
  const int k4 = (gid >> 5) & 63;
  const int ct = gid >> 11;
  if (ct >= NOUT / 16) return;
  const int n = lane & 15;
  const int koff = (lane < 16) ? 0 : 2;
  const int k = k4 * 4 + koff;
  v2f b;
  b.x = W[(size_t)k * NOUT + ct * 16 + n];
  b.y = W[(size_t)(k + 1) * NOUT + ct * 16 + n];
  *(v2f*)(Wt + ((size_t)(ct * 64 + k4)) * 64 + lane * 2) = b;
}

// ---------------------------------------------------------------------------
// 4) GEMM  D[M,NOUT] = A[M,256] @ W[256,NOUT]   via V_WMMA_F32_16X16X4_F32.
//    Block = 256 threads (8 waves) = one 32-row stripe staged in LDS
//    (stride 260 kills the 16-way bank conflict of a 256-dword row stride).
//    Each wave: 2 row-tiles x NT col-tiles; B fragment reused across the two
//    row-tiles -> per K-step: 2 LDS b64 + NT global b64 + 2*NT wmma.
// ---------------------------------------------------------------------------
template <int NOUT, bool RELU_A>
__global__ __launch_bounds__(256) void gemm_wmma_kernel(
    const float* __restrict__ A, const float* __restrict__ Wt,
    float* __restrict__ D) {
  constexpr int NT = NOUT / 128;     // col tiles per wave (2 for 256, 1 for 128)
  constexpr int SSTR = 260;          // padded LDS row stride (floats)
  __shared__ float sA[32 * SSTR];

  const int tid = threadIdx.x;
  const int rowBase = blockIdx.x * 32;

  // Stage the 32x256 A stripe (16 threads cover one 1KB row, two passes).
#pragma unroll
  for (int half = 0; half < 2; ++half) {
    const int r = half * 16 + (tid >> 4);
    const int c = (tid & 15) * 16;
    const float4* src = (const float4*)(A + (size_t)(rowBase + r) * HID + c);
    float4* dst = (float4*)(&sA[r * SSTR + c]);
#pragma unroll
    for (int i = 0; i < 4; ++i) {
      float4 v = src[i];
      if (RELU_A) {
        v.x = fmaxf(v.x, 0.0f); v.y = fmaxf(v.y, 0.0f);
        v.z = fmaxf(v.z, 0.0f); v.w = fmaxf(v.w, 0.0f);
      }
      dst[i] = v;
    }
  }
  __syncthreads();

  const int waveId = tid >> 5;
  const int lane = tid & 31;
  const int m = lane & 15;                   // row (A) / col (B,D) within tile
  const int koff = (lane < 16) ? 0 : 2;      // K sub-offset per ISA 16x4 layout

  v8f acc[2][NT] = {};
  const float* aRow0 = &sA[m * SSTR];
  const float* aRow1 = &sA[(16 + m) * SSTR];
  const float* wBase[NT];
#pragma unroll
  for (int t = 0; t < NT; ++t)
    wBase[t] = Wt + ((size_t)((waveId * NT + t) * 64)) * 64 + lane * 2;

#pragma unroll 8
  for (int k4 = 0; k4 < 64; ++k4) {
    // A fragments: lanes 0-15 hold (K,K+1), lanes 16-31 hold (K+2,K+3)
    v2f a0 = *(const v2f*)(aRow0 + k4 * 4 + koff);
    v2f a1 = *(const v2f*)(aRow1 + k4 * 4 + koff);
#pragma unroll
    for (int t = 0; t < NT; ++t) {
      v2f b = *(const v2f*)(wBase[t] + k4 * 64);   // coalesced 256B/wave
      acc[0][t] = __builtin_amdgcn_wmma_f32_16x16x4_f32(
          false, a0, false, b, (short)0, acc[0][t], false, false);
      acc[1][t] = __builtin_amdgcn_wmma_f32_16x16x4_f32(
          false, a1, false, b, (short)0, acc[1][t], false, false);
    }
  }

  // D layout: VGPR r -> M=r (lanes 0-15) / M=8+r (lanes 16-31), N = lane%16
#pragma unroll
  for (int r = 0; r < 8; ++r) {
    const int mm = (lane < 16) ? r : (8 + r);
#pragma unroll
    for (int rt = 0; rt < 2; ++rt) {
#pragma unroll
      for (int t = 0; t < NT; ++t) {
        const int colBase = (waveId * NT + t) * 16;
        D[(size_t)(rowBase + rt * 16 + mm) * NOUT + colBase + m] = acc[rt][t][r];
      }
    }
  }
}

// ---------------------------------------------------------------------------
// 5) agg[i,:] = bias + dinv[i]^2 * xw[i,:]   (self loop fused with bias init;
//    fully overwrites the output buffer -> no dependence on poisoned state)
// ---------------------------------------------------------------------------
template <int NOUT>
__global__ __launch_bounds__(256) void self_bias_kernel(
    const float* __restrict__ xw, const float* __restrict__ dinv,
    const float* __restrict__ bias, float* __restrict__ agg) {
  const int gid = blockIdx.x * blockDim.x + threadIdx.x;
  const int node = gid / (NOUT / 4);
  const int c4 = (gid % (NOUT / 4)) * 4;
  if (node >= N_NODES) return;
  const float dn = dinv[node];
  const float sn = dn * dn;
  const float4 xv = *(const float4*)(xw + (size_t)node * NOUT + c4);
  const float4 bv = *(const float4*)(bias + c4);
  float4 o;
  o.x = bv.x + sn * xv.x;  o.y = bv.y + sn * xv.y;
  o.z = bv.z + sn * xv.z;  o.w = bv.w + sn * xv.w;
  *(float4*)(agg + (size_t)node * NOUT + c4) = o;
}

// ---------------------------------------------------------------------------
// 6) Edge scatter: agg[col] += dinv[row]*w*dinv[col] * xw[row]
//    NOUT/4 threads per edge, float4 gather + 4 f32 atomics (no-return).
// ---------------------------------------------------------------------------
template <int NOUT>
__global__ __launch_bounds__(256) void scatter_kernel(
    const int* __restrict__ ei, const float* __restrict__ ew,
    const float* __restrict__ dinv, const float* __restrict__ xw,
    float* __restrict__ agg) {
  constexpr int TPE = NOUT / 4;
  const int gid = blockIdx.x * blockDim.x + threadIdx.x;
  const int e = gid / TPE;
  const int c4 = (gid % TPE) * 4;
  if (e >= N_EDGES) return;
  const int r = ei[e];
  const int c = ei[N_EDGES + e];
  const float norm = dinv[r] * ew[e] * dinv[c];
  const float4 xv = *(const float4*)(xw + (size_t)r * NOUT + c4);
  float* dst = agg + (size_t)c * NOUT + c4;
  atomicAdd(dst + 0, norm * xv.x);
  atomicAdd(dst + 1, norm * xv.y);
  atomicAdd(dst + 2, norm * xv.z);
  atomicAdd(dst + 3, norm * xv.w);
}

// ---------------------------------------------------------------------------
// 7) Final ReLU in place
// ---------------------------------------------------------------------------
__global__ __launch_bounds__(256) void relu_kernel(float* __restrict__ p, int n4) {
  const int i = blockIdx.x * blockDim.x + threadIdx.x;
  if (i < n4) {
    float4 v = ((float4*)p)[i];
    v.x = fmaxf(v.x, 0.0f); v.y = fmaxf(v.y, 0.0f);
    v.z = fmaxf(v.z, 0.0f); v.w = fmaxf(v.w, 0.0f);
    ((float4*)p)[i] = v;
  }
}

// ---------------------------------------------------------------------------
extern "C" void kernel_launch(void* const* d_in, const int* in_sizes, int n_in,
                              void* d_out, int out_size, void* d_ws, size_t ws_size,
                              hipStream_t stream) {
  const int*   x     = (const int*)d_in[0];
  const int*   ei    = (const int*)d_in[1];   // [2, E]: rows then cols
  const float* ew    = (const float*)d_in[2];
  const float* syn   = (const float*)d_in[3];
  const float* lem   = (const float*)d_in[4];
  const float* pos   = (const float*)d_in[5];
  const float* sen   = (const float*)d_in[6];
  const float* gamma = (const float*)d_in[7];
  const float* beta  = (const float*)d_in[8];
  const float* W1    = (const float*)d_in[9];
  const float* b1    = (const float*)d_in[10];
  const float* W2    = (const float*)d_in[11];
  const float* b2    = (const float*)d_in[12];
  float* out = (float*)d_out;

  // Workspace layout (all fully rewritten every call):
  //   [0, hb)          h   [N,256] f32   (reused as xw2 [N,128] after GEMM1)
  //   [hb, 2hb)        xw1 [N,256]
  //   [2hb, 3hb)       agg1[N,256]
  //   [3hb, +512K)     deg/dinv [N]
  //   then Wt1 (256K), Wt2 (128K)  -- WMMA-fragment-ordered weights
  char* ws = (char*)d_ws;
  const size_t hb = (size_t)N_NODES * HID * sizeof(float);
  float* h    = (float*)(ws);
  float* xw1  = (float*)(ws + hb);
  float* agg1 = (float*)(ws + 2 * hb);
  float* dinv = (float*)(ws + 3 * hb);
  float* Wt1  = (float*)(ws + 3 * hb + (size_t)(512 * 1024));
  float* Wt2  = (float*)(ws + 3 * hb + (size_t)(768 * 1024));
  float* xw2  = h;  // h dead after GEMM1

  const int B = 256;

  // Weight repack (independent; L2-resident fragment streams for the GEMMs)
  repack_w_kernel<HID><<<((HID / 16) * 64 * 32 + B - 1) / B, B, 0, stream>>>(W1, Wt1);
  repack_w_kernel<OUTF><<<((OUTF / 16) * 64 * 32 + B - 1) / B, B, 0, stream>>>(W2, Wt2);

  // h = LayerNorm(sum of 4 embedding gathers)
  embed_ln_kernel<<<N_NODES / 8, B, 0, stream>>>(x, syn, lem, pos, sen,
                                                 gamma, beta, h);
  // dinv = rsqrt(1 + segsum(w, col))
  deg_init_kernel<<<(N_NODES + B - 1) / B, B, 0, stream>>>(dinv);
  deg_accum_kernel<<<(N_EDGES + B - 1) / B, B, 0, stream>>>(ei + N_EDGES, ew, dinv);
  dinv_kernel<<<(N_NODES + B - 1) / B, B, 0, stream>>>(dinv);

  // Layer 1: xw1 = h @ W1 ; agg1 = b1 + selfloop + edge scatter
  gemm_wmma_kernel<HID, false><<<N_NODES / 32, B, 0, stream>>>(h, Wt1, xw1);
  self_bias_kernel<HID><<<(N_NODES * (HID / 4) + B - 1) / B, B, 0, stream>>>(
      xw1, dinv, b1, agg1);
  scatter_kernel<HID><<<(N_EDGES * (HID / 4) + B - 1) / B, B, 0, stream>>>(
      ei, ew, dinv, xw1, agg1);

  // Layer 2: xw2 = relu(agg1) @ W2 ; out = b2 + selfloop + scatter ; relu
  gemm_wmma_kernel<OUTF, true><<<N_NODES / 32, B, 0, stream>>>(agg1, Wt2, xw2);
  self_bias_kernel<OUTF><<<(N_NODES * (OUTF / 4) + B - 1) / B, B, 0, stream>>>(
      xw2, dinv, b2, out);
  scatter_kernel<OUTF><<<(N_EDGES * (OUTF / 4) + B - 1) / B, B, 0, stream>>>(
      ei, ew, dinv, xw2, out);
  relu_kernel<<<(N_NODES * (OUTF / 4) + B - 1) / B, B, 0, stream>>>(
      out, N_NODES * (OUTF / 4));
}